// Multi_Dim_GMMLoss_15315853377590
// MI455X (gfx1250) — compile-verified
//
#include <hip/hip_runtime.h>
#include <hip/hip_bf16.h>
#include <math.h>

// Problem constants (from reference): B=512, C=256, K=256, L=2
#define NB   512
#define NC   256
#define NK   256
#define BC_K 65536            // C*K
#define MAT  65536            // K*K

typedef __attribute__((ext_vector_type(2))) float v2f;
typedef __attribute__((ext_vector_type(8))) float v8f;

#if __has_builtin(__builtin_amdgcn_wmma_f32_16x16x4_f32)
#define HAVE_WMMA_F32 1
#endif

#if __has_builtin(__builtin_amdgcn_global_load_async_to_lds_b128)
#define HAVE_ASYNC_LDS 1
// Exact parameter types per hipcc diagnostic:
//   param0: '__attribute__((__vector_size__(4*sizeof(int)))) int __device__ *'  (AS1 = global)
//   param1: expected to be the AS3 (__shared__) analog
typedef __attribute__((__vector_size__(4 * sizeof(int)))) int i4vec;
typedef __attribute__((address_space(1))) i4vec* gptr_b128;
typedef __attribute__((address_space(3))) i4vec* lptr_b128;
#endif

#if __has_builtin(__builtin_amdgcn_s_wait_asynccnt)
#define WAIT_ASYNC(n) __builtin_amdgcn_s_wait_asynccnt(n)
#else
#define WAIT_ASYNC(n) asm volatile("s_wait_asynccnt %0" ::"i"(n) : "memory")
#endif

// ---------------- ws layout (floats) ----------------
// [0..3]          counts0, counts1, safe0, safe1
// [16..527]       logdet[l*256+c]
// [1024..132095]  means (2, 256, 256)
// [132096.. ]     sigma (2, 256, 256, 256)  — label1 overwritten by its inverse
#define OFF_LOGDET 16
#define OFF_MEANS  1024
#define OFF_SIGMA  132096

// ---------------------------------------------------------------- counts
__global__ void gmm_counts(const int* __restrict__ labels, float* __restrict__ ws) {
    __shared__ int cnt[2];
    int tid = threadIdx.x;
    if (tid < 2) cnt[tid] = 0;
    __syncthreads();
    atomicAdd(&cnt[labels[tid] & 1], 1);
    __syncthreads();
    if (tid == 0) {
        ws[0] = (float)cnt[0];
        ws[1] = (float)cnt[1];
        ws[2] = fmaxf((float)cnt[0], 1.0f);
        ws[3] = fmaxf((float)cnt[1], 1.0f);
    }
}

// ---------------------------------------------------------------- means
__global__ __launch_bounds__(256) void gmm_means(const float* __restrict__ x,
                                                 const int* __restrict__ labels,
                                                 const float* __restrict__ ws,
                                                 float* __restrict__ means) {
    __shared__ int sLab[NB];
    int tid = threadIdx.x;
    sLab[tid]       = labels[tid];
    sLab[tid + 256] = labels[tid + 256];
    __syncthreads();
    int cid = blockIdx.x * 256 + tid;          // 0..65535 == c*256+k
    float s0 = 0.f, s1 = 0.f;
    for (int b = 0; b < NB; ++b) {
        float v = x[(size_t)b * BC_K + cid];
        if (sLab[b] == 0) s0 += v; else s1 += v;
    }
    means[cid]        = s0 / ws[2];
    means[BC_K + cid] = s1 / ws[3];
}

// ---------------------------------------------------------------- raw-Gram WMMA kernel
// One block per channel c. Computes G_all = X^T X and G0 = (mask0 . X)^T X over
// the 512-row batch with f32 WMMA (K-dim = batch, stepped by 4).
//   Sigma[0] = cov + G0/safe0            - mu0 mu0^T
//   Sigma[1] = cov + (G_all - G0)/safe1  - mu1 mu1^T
// Batch chunks are double-buffered in LDS via async global->LDS b128 copies.
#define CHG     32            // batch rows per chunk
#define LSTR_G  260           // padded row stride (floats): 16B-aligned rows, conflict-free

__device__ __forceinline__ void issue_chunk(const float* xc, float* lbuf, int b0, int tid) {
#ifdef HAVE_ASYNC_LDS
    #pragma unroll
    for (int i = 0; i < 8; ++i) {
        int f4  = tid + i * 256;              // 0..2047 float4s of the 32x256 chunk
        int row = f4 >> 6;
        int c4  = f4 & 63;
        const float* g = xc + (size_t)(b0 + row) * BC_K + c4 * 4;
        float*       l = lbuf + row * LSTR_G + c4 * 4;
        __builtin_amdgcn_global_load_async_to_lds_b128((gptr_b128)g, (lptr_b128)l, 0, 0);
    }
#else
    #pragma unroll
    for (int i = 0; i < 8; ++i) {
        int f4  = tid + i * 256;
        int row = f4 >> 6;
        int c4  = f4 & 63;
        float4 v = *(const float4*)(xc + (size_t)(b0 + row) * BC_K + c4 * 4);
        *(float4*)(lbuf + row * LSTR_G + c4 * 4) = v;
    }
#endif
}

__global__ __launch_bounds__(256) void gmm_scatter_sigma(const float* __restrict__ x,
                                                         const float* __restrict__ cov,
                                                         const float* __restrict__ means,
                                                         const int* __restrict__ labels,
                                                         const float* __restrict__ ws,
                                                         float* __restrict__ sigma) {
    extern __shared__ float sBuf[];            // 2 * CHG * LSTR_G floats (double buffer)
    __shared__ float sMean0[NK], sMean1[NK];
    __shared__ unsigned sMask[NB / CHG];       // per-chunk label-0 bitmask

    const int c    = blockIdx.x;
    const int tid  = threadIdx.x;
    const int wave = tid >> 5;
    const int lane = tid & 31;
    const int half = lane >> 4;                // 0: lanes 0-15, 1: lanes 16-31
    const int l16  = lane & 15;

    sMean0[tid] = means[c * NK + tid];
    sMean1[tid] = means[BC_K + c * NK + tid];
    if (tid < NB / CHG) {
        unsigned mb = 0;
        for (int r = 0; r < CHG; ++r)
            mb |= (unsigned)(labels[tid * CHG + r] == 0) << r;
        sMask[tid] = mb;
    }
    __syncthreads();

    const float inv_s0 = 1.0f / ws[2];
    const float inv_s1 = 1.0f / ws[3];
    const float* xc = x + c * NK;              // element (b,k) at xc[b*BC_K + k]
    float* out0 = sigma + (size_t)c * MAT;
    float* out1 = sigma + (size_t)(NC + c) * MAT;

    for (int pass = 0; pass < 8; ++pass) {
        v8f acc0[4], accA[4];
        #pragma unroll
        for (int q = 0; q < 4; ++q) {
            acc0[q] = (v8f){0.f,0.f,0.f,0.f,0.f,0.f,0.f,0.f};
            accA[q] = (v8f){0.f,0.f,0.f,0.f,0.f,0.f,0.f,0.f};
        }
        const int tbase = pass * 32 + wave * 4;

        issue_chunk(xc, sBuf, 0, tid);
        for (int chunk = 0; chunk < NB / CHG; ++chunk) {
            if (chunk + 1 < NB / CHG) {
                issue_chunk(xc, sBuf + ((chunk + 1) & 1) * (CHG * LSTR_G),
                            (chunk + 1) * CHG, tid);
#ifdef HAVE_ASYNC_LDS
                WAIT_ASYNC(8);                 // previous chunk's 8 copies done
#endif
            } else {
#ifdef HAVE_ASYNC_LDS
                WAIT_ASYNC(0);
#endif
            }
            __syncthreads();

            const float* sM = sBuf + (chunk & 1) * (CHG * LSTR_G);
            const unsigned mb = sMask[chunk];

            #pragma unroll
            for (int q = 0; q < 4; ++q) {
                const int t  = tbase + q;
                const int ti = t >> 4;
                const int tj = t & 15;
                #pragma unroll
                for (int bb = 0; bb < CHG; bb += 4) {
                    // A (16x4): VGPR0={K0|K2}, VGPR1={K1|K3} over lane halves
                    float ax = sM[(bb + 2 * half + 0) * LSTR_G + ti * 16 + l16];
                    float ay = sM[(bb + 2 * half + 1) * LSTR_G + ti * 16 + l16];
                    v2f a, a0, bv;
                    a.x = ax; a.y = ay;
                    a0.x = ((mb >> (bb + 2 * half + 0)) & 1u) ? ax : 0.0f;
                    a0.y = ((mb >> (bb + 2 * half + 1)) & 1u) ? ay : 0.0f;
                    // B (4x16): row-striped across lane halves per VGPR
                    bv.x = sM[(bb + half    ) * LSTR_G + tj * 16 + l16];
                    bv.y = sM[(bb + 2 + half) * LSTR_G + tj * 16 + l16];
#ifdef HAVE_WMMA_F32
                    accA[q] = __builtin_amdgcn_wmma_f32_16x16x4_f32(
                        false, a,  false, bv, (short)0, accA[q], false, false);
                    acc0[q] = __builtin_amdgcn_wmma_f32_16x16x4_f32(
                        false, a0, false, bv, (short)0, acc0[q], false, false);
#else
                    #pragma unroll
                    for (int r = 0; r < 8; ++r) {
                        for (int kk = 0; kk < 4; ++kk) {
                            float am = sM[(bb + kk) * LSTR_G + ti * 16 + r + 8 * half];
                            float bm = sM[(bb + kk) * LSTR_G + tj * 16 + l16];
                            accA[q][r] = fmaf(am, bm, accA[q][r]);
                            float m0 = ((mb >> (bb + kk)) & 1u) ? am : 0.0f;
                            acc0[q][r] = fmaf(m0, bm, acc0[q][r]);
                        }
                    }
#endif
                }
            }
            __syncthreads();
        }

        // store: D layout — VGPR r holds M=r (lanes 0-15) / M=r+8 (lanes 16-31), N=l16
        #pragma unroll
        for (int q = 0; q < 4; ++q) {
            const int t  = tbase + q;
            const int ti = t >> 4;
            const int tj = t & 15;
            const int gc = tj * 16 + l16;
            #pragma unroll
            for (int r = 0; r < 8; ++r) {
                const int gr = ti * 16 + r + 8 * half;
                float g0 = acc0[q][r];
                float ga = accA[q][r];
                float cv = cov[gr * NK + gc];
                out0[gr * NK + gc] = cv + g0 * inv_s0        - sMean0[gr] * sMean0[gc];
                out1[gr * NK + gc] = cv + (ga - g0) * inv_s1 - sMean1[gr] * sMean1[gc];
            }
        }
    }
}

// ---------------------------------------------------------------- in-LDS Gauss-Jordan inverse + logdet
// One 256-thread block per matrix; full 256x257 fp32 matrix in 257 KB LDS
// (possible only with CDNA5's 320 KB/WGP LDS). SPD => no pivoting.
#define LSTRIDE 257
__global__ __launch_bounds__(256) void gmm_invert(float* __restrict__ sigma,
                                                  float* __restrict__ logdet) {
    extern __shared__ float A[];          // 256*257 floats
    const int bid = blockIdx.x;           // l*256 + c
    const int l   = bid >> 8;
    const int tid = threadIdx.x;
    float* g = sigma + (size_t)bid * MAT;

    for (int idx = tid; idx < MAT; idx += 256)
        A[(idx >> 8) * LSTRIDE + (idx & 255)] = g[idx];

    float ld = 0.0f;
    for (int k = 0; k < NK; ++k) {
        __syncthreads();
        float p    = A[k * LSTRIDE + k];
        float invp = 1.0f / p;
        if (tid == 0) ld += logf(p);
        if (tid != k) A[k * LSTRIDE + tid] *= invp;     // scale pivot row (j != k)
        __syncthreads();
        if (tid != k) {                                  // eliminate row `tid`
            float f = A[tid * LSTRIDE + k];
            for (int j = 0; j < NK; ++j) {
                if (j != k)
                    A[tid * LSTRIDE + j] = fmaf(-f, A[k * LSTRIDE + j], A[tid * LSTRIDE + j]);
            }
            A[tid * LSTRIDE + k] = -f * invp;            // pivot column
        } else {
            A[k * LSTRIDE + k] = invp;                   // nobody reads A[k][k] this phase
        }
    }
    __syncthreads();

    if (tid == 0) logdet[bid] = ld;
    if (l == 1) {                                        // keep Sigma[0]; store Sinv[1]
        for (int idx = tid; idx < MAT; idx += 256)
            g[idx] = A[(idx >> 8) * LSTRIDE + (idx & 255)];
    }
}

// ---------------------------------------------------------------- final KL reduction
__global__ void gmm_zero(float* __restrict__ out) { if (threadIdx.x == 0) out[0] = 0.0f; }

__global__ __launch_bounds__(256) void gmm_kl(const float* __restrict__ sigma,
                                              const float* __restrict__ means,
                                              const float* __restrict__ logdet,
                                              const float* __restrict__ ws,
                                              float* __restrict__ out) {
    __shared__ float sd[NK];
    __shared__ float redT[256];
    __shared__ float redQ[256];
    const int c   = blockIdx.x;
    const int tid = threadIdx.x;
    sd[tid] = means[BC_K + c * NK + tid] - means[c * NK + tid];
    __syncthreads();

    const float* S0 = sigma + (size_t)c * MAT;           // Sigma[0,c] (symmetric)
    const float* Si = sigma + (size_t)(NC + c) * MAT;    // Sinv[1,c]
    float tr = 0.f, quad = 0.f;
    for (int idx = tid; idx < MAT; idx += 256) {         // coalesced over columns
        float si = Si[idx];
        tr   = fmaf(si, S0[idx], tr);                    // tr(Sinv1 * Sigma0), S0 symmetric
        quad = fmaf(sd[idx >> 8] * si, sd[idx & 255], quad);
    }
    redT[tid] = tr; redQ[tid] = quad;
    __syncthreads();
    for (int s = 128; s > 0; s >>= 1) {
        if (tid < s) { redT[tid] += redT[tid + s]; redQ[tid] += redQ[tid + s]; }
        __syncthreads();
    }
    if (tid == 0) {
        float kl = 0.5f * (redT[0] + redQ[0] - (float)NK + logdet[NC + c] - logdet[c]);
        float scale = ws[0] * ws[1] * (1.0f / ((float)NC * (float)NB * (float)NB));
        atomicAdd(out, kl * scale);
    }
}

// ---------------------------------------------------------------- launcher
extern "C" void kernel_launch(void* const* d_in, const int* in_sizes, int n_in,
                              void* d_out, int out_size, void* d_ws, size_t ws_size,
                              hipStream_t stream) {
    const float* mu     = (const float*)d_in[0];   // (512,256,16,16) fp32
    const float* cov    = (const float*)d_in[1];   // (256,256) fp32
    const int*   labels = (const int*)d_in[2];     // (512,) int
    float* out = (float*)d_out;
    float* ws  = (float*)d_ws;

    float* means  = ws + OFF_MEANS;
    float* lgd    = ws + OFF_LOGDET;
    float* sigma  = ws + OFF_SIGMA;   // needs ~135 MB of workspace

    gmm_counts<<<1, 512, 0, stream>>>(labels, ws);
    gmm_means<<<BC_K / 256, 256, 0, stream>>>(mu, labels, ws, means);
    gmm_scatter_sigma<<<NC, 256, 2 * CHG * LSTR_G * sizeof(float), stream>>>(
        mu, cov, means, labels, ws, sigma);
    gmm_invert<<<2 * NC, 256, NK * LSTRIDE * sizeof(float), stream>>>(sigma, lgd);
    gmm_zero<<<1, 64, 0, stream>>>(out);
    gmm_kl<<<NC, 256, 0, stream>>>(sigma, means, lgd, ws, out);
}